// GemoLoss_352187318694
// MI455X (gfx1250) — compile-verified
//
#include <hip/hip_runtime.h>
#include <hip/hip_bf16.h>
#include <math.h>

// ---------------- geometry constants ----------------
#define HD 384
#define WD 384
#define KP 25               // kernel pad / max radius
#define NPLANE 6            // B*3 = 2*3
#define PLANE_ELEMS (HD*WD) // 147456
#define TX 32
#define TY 16
#define NTHREADS (TX*TY)    // 512
#define GX (WD/TX)          // 12
#define GY (HD/TY)          // 24
#define NBLOCKS (GX*GY*NPLANE) // 1728 = 27*64 (exact multiple of 64 for wmma reduce)

// LDS tile: rows [y0-25, y0+TY+25), cols aligned to 4 elems: [x0-28, x0+60)
#define LROWS  (TY + 2*KP)      // 66
#define XOFF   28               // left halo (25 rounded up to multiple of 4)
#define NCHUNK 22               // 22 x 4 floats = 88 cols  (covers x0-28 .. x0+59)
#define LCOLS  (NCHUNK*4)       // 88
#define LSTR   92               // padded stride: 368 B/row (16B-aligned rows)

typedef float v2f __attribute__((ext_vector_type(2)));
typedef float v8f __attribute__((ext_vector_type(8)));
typedef int   gv4i __attribute__((vector_size(16)));   // matches b128 builtin param

#define AS1 __attribute__((address_space(1)))
#define AS3 __attribute__((address_space(3)))

// ---------------- CDNA5 async global->LDS copies ----------------
#if defined(__has_builtin)
#  if __has_builtin(__builtin_amdgcn_global_load_async_to_lds_b32)
#    define HAVE_ASYNC_LDS32 1
#  endif
#  if __has_builtin(__builtin_amdgcn_global_load_async_to_lds_b128)
#    define HAVE_ASYNC_LDS128 1
#  endif
#  if __has_builtin(__builtin_amdgcn_wmma_f32_16x16x4_f32)
#    define HAVE_WMMA_F32 1
#  endif
#endif

// copy 4 consecutive floats (16B-aligned both sides), tracked with ASYNCcnt
__device__ __forceinline__ void async_copy_4xf32(float* lds_dst, const float* gsrc)
{
#if defined(HAVE_ASYNC_LDS128)
    __builtin_amdgcn_global_load_async_to_lds_b128(
        (AS1 gv4i*)(void*)gsrc, (AS3 gv4i*)(void*)lds_dst, 0, 0);
#elif defined(HAVE_ASYNC_LDS32)
#warning "b128 async builtin unavailable -- using 4x b32 async"
    __builtin_amdgcn_global_load_async_to_lds_b32(
        (AS1 int*)(void*)gsrc, (AS3 int*)(void*)lds_dst, 0, 0);
    __builtin_amdgcn_global_load_async_to_lds_b32(
        (AS1 int*)(void*)(gsrc+1), (AS3 int*)(void*)(lds_dst+1), 0, 0);
    __builtin_amdgcn_global_load_async_to_lds_b32(
        (AS1 int*)(void*)(gsrc+2), (AS3 int*)(void*)(lds_dst+2), 0, 0);
    __builtin_amdgcn_global_load_async_to_lds_b32(
        (AS1 int*)(void*)(gsrc+3), (AS3 int*)(void*)(lds_dst+3), 0, 0);
#else
#warning "async to-LDS builtins unavailable -- direct copy fallback"
    lds_dst[0] = gsrc[0]; lds_dst[1] = gsrc[1];
    lds_dst[2] = gsrc[2]; lds_dst[3] = gsrc[3];
#endif
}

__device__ __forceinline__ void wait_async_all()
{
#if defined(HAVE_ASYNC_LDS128) || defined(HAVE_ASYNC_LDS32)
#  if defined(__has_builtin) && __has_builtin(__builtin_amdgcn_s_wait_asynccnt)
    __builtin_amdgcn_s_wait_asynccnt(0);
#  else
    asm volatile("s_wait_asynccnt 0" ::: "memory");
#  endif
#endif
}

// ---------------- WMMA f32 16x16x4 wave-wide sum ----------------
// Sums n64*64 consecutive floats from src. Entire wave must be active
// (EXEC all ones -- ISA requirement for WMMA). Deterministic fixed order.
// D = A x ones + C accumulates row sums of A; every column of D is identical;
// lanes 0/16 hold rows 0-7 / 8-15 of column 0.
__device__ __forceinline__ float wave_wmma_sum(const float* __restrict__ src, int n64)
{
    const int lane = threadIdx.x & 31;
#ifdef HAVE_WMMA_F32
    const int m     = lane & 15;
    const int khalf = (lane >> 4) << 1;     // lanes 0-15 -> K=0,1 ; lanes 16-31 -> K=2,3
    v2f bv; bv[0] = 1.0f; bv[1] = 1.0f;     // B = ones(4x16)
    v8f cv = {0.f,0.f,0.f,0.f,0.f,0.f,0.f,0.f};
    for (int t = 0; t < n64; ++t) {
        const int base = t*64 + m*4 + khalf;   // bijective (m,k) -> linear map
        v2f av; av[0] = src[base]; av[1] = src[base+1];
        cv = __builtin_amdgcn_wmma_f32_16x16x4_f32(
                 false, av, false, bv, (short)0, cv, false, false);
    }
    float s = cv[0]+cv[1]+cv[2]+cv[3]+cv[4]+cv[5]+cv[6]+cv[7];
    return __shfl(s, 0, 32) + __shfl(s, 16, 32);
#else
#warning "wmma_f32_16x16x4_f32 builtin unavailable -- scalar reduction fallback"
    float s = 0.f;
    for (int t = 0; t < n64; ++t)
        for (int i = lane; i < 64; i += 32) s += src[t*64 + i];
    for (int off = 16; off > 0; off >>= 1) s += __shfl_xor(s, off, 32);
    return s;
#endif
}

// ---------------- stage 1: fused sparse gemo-diff ----------------
// diff = left_gemo - right_gemo, using at most 2 adjacent kernel radii/pixel.
__global__ __launch_bounds__(NTHREADS)
void gemo_diff_kernel(const float* __restrict__ blur_map,
                      const float* __restrict__ x,
                      const float* __restrict__ kernels,
                      float* __restrict__ diff)
{
    __shared__ __attribute__((aligned(16))) float sL[LROWS*LSTR];
    __shared__ __attribute__((aligned(16))) float sR[LROWS*LSTR];

    const int tid = threadIdx.x;
    const int tx  = tid & (TX-1);
    const int ty  = tid >> 5;
    const int x0  = blockIdx.x * TX;
    const int y0  = blockIdx.y * TY;
    const int pl  = blockIdx.z;            // b*3 + c
    const int b   = pl / 3, c = pl % 3;

    const float* xL = x + (size_t)(b*6 + c    ) * PLANE_ELEMS;
    const float* xR = x + (size_t)(b*6 + 3 + c) * PLANE_ELEMS;

    // zero LDS (covers zero-padded conv halo), then async-load in-bounds chunks
    for (int e = tid; e < LROWS*LSTR; e += NTHREADS) { sL[e] = 0.f; sR[e] = 0.f; }
    __syncthreads();
    // 4-elem aligned chunks: each chunk is entirely inside or outside the image
    for (int e = tid; e < LROWS*NCHUNK; e += NTHREADS) {
        const int row = e / NCHUNK, ch = e - row*NCHUNK;
        const int gy = y0 + row - KP;
        const int gx = x0 - XOFF + ch*4;              // multiple of 4
        if ((unsigned)gy < (unsigned)HD && (unsigned)gx <= (unsigned)(WD-4)) {
            const int g = gy*WD + gx;                 // 16B-aligned global offset
            const int l = row*LSTR + ch*4;            // 16B-aligned LDS offset
            async_copy_4xf32(&sL[l], &xL[g]);
            async_copy_4xf32(&sR[l], &xR[g]);
        }
    }
    wait_async_all();
    __syncthreads();

    const int yy = y0 + ty, xx = x0 + tx;
    const float u = blur_map[(size_t)pl*PLANE_ELEMS + yy*WD + xx];
    const float v = fabsf(u);
    float acc = 0.f;

    if (v < 25.f) {
        const int   ilo  = (int)v;                 // floor (v >= 0)
        const float frac = v - (float)ilo;
        const bool  hiv  = (frac > 0.f) && (ilo < 24);
        const int   r    = hiv ? (ilo + 1) : ilo;  // loop radius (zeros outside support)
        const float wl   = 1.f - frac;
        const float wh   = hiv ? frac : 0.f;

        float wp_lo, wp_hi, wn_lo, wn_hi;          // pos/neg weights at ilo / ilo+1
        if (u >= 0.f) { wp_lo = wl; wp_hi = wh; wn_lo = (v < 1.f) ? wl : 0.f; wn_hi = 0.f; }
        else          { wn_lo = wl; wn_hi = wh; wp_lo = (v < 1.f) ? wl : 0.f; wp_hi = 0.f; }
        if (ilo == 0) { wp_lo *= 0.5f; wn_lo *= 0.5f; }   // half0

        const float* klo = kernels + (size_t)ilo * (51*51);
        const float* khi = kernels + (size_t)(hiv ? (ilo+1) : ilo) * (51*51);
        const int lc = XOFF + tx;                  // center col in LDS

        // per tap (oy,ox): k*(pos*(xL[+ox]-xR[-ox]) + neg*(xL[-ox]-xR[+ox]))
        for (int oy = -r; oy <= r; ++oy) {
            const float* kr   = klo + (KP+oy)*51 + KP;
            const float* kr2  = khi + (KP+oy)*51 + KP;
            const float* rowL = sL + (ty + KP + oy)*LSTR + lc;
            const float* rowR = sR + (ty + KP + oy)*LSTR + lc;
            for (int ox = -r; ox <= r; ++ox) {
                const float kl = kr[ox];
                const float kh = kr2[ox];
                const float kp = wp_lo*kl + wp_hi*kh;
                const float kn = wn_lo*kl + wn_hi*kh;
                acc = fmaf(kp, rowL[ox]  - rowR[-ox], acc);
                acc = fmaf(kn, rowL[-ox] - rowR[ox],  acc);
            }
        }
    }
    diff[(size_t)pl*PLANE_ELEMS + yy*WD + xx] = acc;
}

// ---------------- stage 2: 3x3 reflect gaussian + |.| + block sums ----------------
__global__ __launch_bounds__(NTHREADS)
void gauss_abs_sum_kernel(const float* __restrict__ diff, float* __restrict__ partials)
{
    const int tid = threadIdx.x;
    const int tx  = tid & (TX-1), ty = tid >> 5;
    const int xx  = blockIdx.x*TX + tx, yy = blockIdx.y*TY + ty;
    const int pl  = blockIdx.z;

    const float a   = expf(-1.0f / 4.5f);
    const float inv = 1.f / (1.f + 2.f*a);
    const float w[3] = { a*inv, inv, a*inv };

    const float* d = diff + (size_t)pl*PLANE_ELEMS;
    float acc = 0.f;
    #pragma unroll
    for (int dy = -1; dy <= 1; ++dy) {
        int sy = yy + dy; sy = sy < 0 ? -sy : (sy >= HD ? 2*HD - 2 - sy : sy);
        float ra = 0.f;
        #pragma unroll
        for (int dx = -1; dx <= 1; ++dx) {
            int sx = xx + dx; sx = sx < 0 ? -sx : (sx >= WD ? 2*WD - 2 - sx : sx);
            ra = fmaf(w[dx+1], d[sy*WD + sx], ra);
        }
        acc = fmaf(w[dy+1], ra, acc);
    }

    __shared__ float red[NTHREADS];
    red[tid] = fabsf(acc);
    __syncthreads();
    const float s = wave_wmma_sum(red, NTHREADS/64);   // all waves full-EXEC
    if (tid == 0)
        partials[(blockIdx.z*gridDim.y + blockIdx.y)*gridDim.x + blockIdx.x] = s;
}

// ---------------- stage 3: TV block sums ----------------
__global__ __launch_bounds__(NTHREADS)
void tv_sum_kernel(const float* __restrict__ bm, float* __restrict__ partials)
{
    const int tid = threadIdx.x;
    const int tx  = tid & (TX-1), ty = tid >> 5;
    const int xx  = blockIdx.x*TX + tx, yy = blockIdx.y*TY + ty;
    const int pl  = blockIdx.z;

    const float* p = bm + (size_t)pl*PLANE_ELEMS;
    const float v0 = p[yy*WD + xx];
    float acc = 0.f;
    if (yy < HD-1) acc += fabsf(p[(yy+1)*WD + xx] - v0);
    if (xx < WD-1) acc += fabsf(p[yy*WD + xx + 1] - v0);

    __shared__ float red[NTHREADS];
    red[tid] = acc;
    __syncthreads();
    const float s = wave_wmma_sum(red, NTHREADS/64);
    if (tid == 0)
        partials[(blockIdx.z*gridDim.y + blockIdx.y)*gridDim.x + blockIdx.x] = s;
}

// ---------------- stage 4: single-wave WMMA finalize ----------------
__global__ __launch_bounds__(32)
void finalize_kernel(const float* __restrict__ p2, const float* __restrict__ p3,
                     float* __restrict__ out)
{
    const float s2 = wave_wmma_sum(p2, NBLOCKS/64);   // 1728 = 27*64
    const float s3 = wave_wmma_sum(p3, NBLOCKS/64);
    if (threadIdx.x == 0) {
        const float unsup = s2 * (1000.0f / (float)(NPLANE*PLANE_ELEMS));
        out[0] = unsup;
        out[1] = s3 * 1e-5f;
        out[2] = unsup;
    }
}

// ---------------- host launcher ----------------
extern "C" void kernel_launch(void* const* d_in, const int* in_sizes, int n_in,
                              void* d_out, int out_size, void* d_ws, size_t ws_size,
                              hipStream_t stream)
{
    (void)in_sizes; (void)n_in; (void)out_size; (void)ws_size;
    const float* blur_map = (const float*)d_in[0];   // (2,3,384,384) f32
    const float* x        = (const float*)d_in[1];   // (2,6,384,384) f32
    const float* kernels  = (const float*)d_in[2];   // (25,51,51)   f32
    float* out = (float*)d_out;

    float* diff = (float*)d_ws;                       // 6*384*384 floats (~3.5 MB)
    float* p2   = diff + (size_t)NPLANE*PLANE_ELEMS;  // 1728 floats
    float* p3   = p2 + NBLOCKS;                       // 1728 floats

    dim3 grid(GX, GY, NPLANE), blk(NTHREADS);
    gemo_diff_kernel   <<<grid, blk, 0, stream>>>(blur_map, x, kernels, diff);
    gauss_abs_sum_kernel<<<grid, blk, 0, stream>>>(diff, p2);
    tv_sum_kernel      <<<grid, blk, 0, stream>>>(blur_map, p3);
    finalize_kernel    <<<dim3(1), dim3(32), 0, stream>>>(p2, p3, out);
}